// ODEFunc_3435973837309
// MI455X (gfx1250) — compile-verified
//
#include <hip/hip_runtime.h>
#include <hip/hip_bf16.h>

#define GAMMA 0.5f
#define WPB 8  // waves per 256-thread block (wave32)

// ---------------------------------------------------------------------------
// Kernel 1: out[0:nd) = -GAMMA * h ; out[nd:total) = 0
// ---------------------------------------------------------------------------
__global__ void ode_init_kernel(const float* __restrict__ x,
                                float* __restrict__ out,
                                int nd, int total) {
    const int gid = blockIdx.x * blockDim.x + threadIdx.x;
    const int stride = gridDim.x * blockDim.x;
    const int nd4 = nd >> 2;
    const int tot4 = total >> 2;
    const float4* __restrict__ x4 = (const float4*)x;
    float4* __restrict__ o4 = (float4*)out;
    for (int i = gid; i < tot4; i += stride) {
        if (i < nd4) {
            float4 v = x4[i];
            float4 r;
            r.x = -GAMMA * v.x; r.y = -GAMMA * v.y;
            r.z = -GAMMA * v.z; r.w = -GAMMA * v.w;
            o4[i] = r;
        } else {
            float4 z; z.x = 0.f; z.y = 0.f; z.z = 0.f; z.w = 0.f;
            o4[i] = z;
        }
    }
    // scalar tail in case total % 4 != 0
    for (int i = (tot4 << 2) + gid; i < total; i += stride) {
        out[i] = (i < nd) ? -GAMMA * x[i] : 0.0f;
    }
}

// ---------------------------------------------------------------------------
// Kernel 2: scatter-add.  One wave handles 32 edges per chunk; edge params
// (src, dst, e) are staged into per-wave LDS buffers with CDNA5 async
// global->LDS DMA (ASYNCcnt), double buffered.  Each lane owns 2 features.
// Scatter uses native no-return global_atomic_add_f32 at device scope.
// ---------------------------------------------------------------------------
__global__ void ode_scatter_kernel(const float* __restrict__ x,
                                   const int* __restrict__ src,
                                   const int* __restrict__ dst,
                                   float* __restrict__ out,
                                   int E, int nd) {
    const float* __restrict__ h  = x;       // [N, 64] node features
    const float* __restrict__ ew = x + nd;  // [E] edge weights

    // per wave: 2 buffers x 32 edges x {src, dst, e} = 768 B
    __shared__ unsigned stage[WPB][2][32 * 3];

    const int lane = threadIdx.x & 31;
    const int wid  = threadIdx.x >> 5;
    const int wgl  = blockIdx.x * WPB + wid;      // global wave id
    const int wstr = gridDim.x * WPB;             // wave-grid stride
    const int nchunks = (E + 31) >> 5;

    // LDS byte offset of this lane's 12-byte slot in each buffer.
    // (flat address of LDS carries the LDS offset in its low 32 bits)
    const unsigned l0 = (unsigned)(unsigned long long)&stage[wid][0][lane * 3];
    const unsigned l1 = (unsigned)(unsigned long long)&stage[wid][1][lane * 3];

    const unsigned lane8 = (unsigned)lane << 3;   // byte offset of lane's float2

    int c = wgl;
    if (c >= nchunks) return;

    // --- prologue: async-stage chunk c into buffer 0 ---
    {
        const int eidx = (c << 5) + lane;
        if (eidx < E) {
            const unsigned voff = (unsigned)eidx << 2;   // byte offset into arrays
            asm volatile("global_load_async_to_lds_b32 %0, %1, %2 offset:0"
                         :: "v"(l0),      "v"(voff), "s"(src) : "memory");
            asm volatile("global_load_async_to_lds_b32 %0, %1, %2 offset:0"
                         :: "v"(l0 + 4u), "v"(voff), "s"(dst) : "memory");
            asm volatile("global_load_async_to_lds_b32 %0, %1, %2 offset:0"
                         :: "v"(l0 + 8u), "v"(voff), "s"(ew)  : "memory");
        }
    }

    int p = 0;  // buffer currently holding chunk c
    for (; c < nchunks; c += wstr, p ^= 1) {
        const int cn = c + wstr;
        if (cn < nchunks) {
            // async-stage the NEXT chunk into the other buffer, then wait for
            // the current chunk's 3 DMAs (loads complete in order: <=3 left
            // outstanding means the older 3 are done).
            const unsigned lb = p ? l0 : l1;
            const int eidx = (cn << 5) + lane;
            if (eidx < E) {
                const unsigned voff = (unsigned)eidx << 2;
                asm volatile("global_load_async_to_lds_b32 %0, %1, %2 offset:0"
                             :: "v"(lb),      "v"(voff), "s"(src) : "memory");
                asm volatile("global_load_async_to_lds_b32 %0, %1, %2 offset:0"
                             :: "v"(lb + 4u), "v"(voff), "s"(dst) : "memory");
                asm volatile("global_load_async_to_lds_b32 %0, %1, %2 offset:0"
                             :: "v"(lb + 8u), "v"(voff), "s"(ew)  : "memory");
            }
            asm volatile("s_wait_asynccnt 0x3" ::: "memory");
        } else {
            asm volatile("s_wait_asynccnt 0x0" ::: "memory");
        }

        // --- consume chunk c from LDS (broadcast reads, wave-uniform) ---
        const int nvalid = min(32, E - (c << 5));
        #pragma unroll 4
        for (int k = 0; k < nvalid; ++k) {
            const int   s = (int)stage[wid][p][k * 3 + 0];
            const int   d = (int)stage[wid][p][k * 3 + 1];
            const float w = __uint_as_float(stage[wid][p][k * 3 + 2]);
            // lane owns features 2*lane, 2*lane+1 : coalesced 256B/edge, L2-hot
            const float2 hv = *(const float2*)(h + ((size_t)s << 6) + (lane << 1));
            const float m0 = hv.x * w;
            const float m1 = hv.y * w;
            // out byte offset for this lane's pair: d*256 + lane*8 (< 2^32)
            const unsigned ooff = ((unsigned)d << 8) + lane8;
            asm volatile("global_atomic_add_f32 %0, %1, %2 offset:0 scope:SCOPE_DEV"
                         :: "v"(ooff), "v"(m0), "s"(out) : "memory");
            asm volatile("global_atomic_add_f32 %0, %1, %2 offset:4 scope:SCOPE_DEV"
                         :: "v"(ooff), "v"(m1), "s"(out) : "memory");
        }
    }
}

// ---------------------------------------------------------------------------
// Inputs (setup_inputs order): t[1], x[N*D+E], src[E], dst[E]
// Output: [N*D + E] float32
// ---------------------------------------------------------------------------
extern "C" void kernel_launch(void* const* d_in, const int* in_sizes, int n_in,
                              void* d_out, int out_size, void* d_ws, size_t ws_size,
                              hipStream_t stream) {
    const float* x   = (const float*)d_in[1];
    const int*   src = (const int*)d_in[2];
    const int*   dst = (const int*)d_in[3];
    float* out = (float*)d_out;

    const int E     = in_sizes[2];
    const int nd    = in_sizes[1] - E;   // N * 64
    const int total = out_size;          // nd + E

    int tot4 = total >> 2;
    int b1 = (tot4 + 255) / 256;
    if (b1 > 2048) b1 = 2048;
    if (b1 < 1) b1 = 1;
    ode_init_kernel<<<b1, 256, 0, stream>>>(x, out, nd, total);

    const int nchunks = (E + 31) / 32;
    int b2 = (nchunks + WPB - 1) / WPB;
    if (b2 > 2048) b2 = 2048;
    if (b2 < 1) b2 = 1;
    ode_scatter_kernel<<<b2, 256, 0, stream>>>(x, src, dst, out, E, nd);
}